// simpleGCN_67645734912987
// MI455X (gfx1250) — compile-verified
//
#include <hip/hip_runtime.h>
#include <math.h>

typedef float v2f __attribute__((ext_vector_type(2)));
typedef float v8f __attribute__((ext_vector_type(8)));

#define HDIM 128

// ---------------- zero fill ----------------
__global__ void k_zero(float* __restrict__ p, int n) {
  int i = blockIdx.x * blockDim.x + threadIdx.x;
  if (i < n) p[i] = 0.0f;
}

// ---------------- edge scatter-add (generic feature count) ----------------
__global__ void k_edge_scatter(const float* __restrict__ hin,
                               const int* __restrict__ src,
                               const int* __restrict__ dst,
                               float* __restrict__ agg,
                               int total, int Kf) {
  int idx = blockIdx.x * blockDim.x + threadIdx.x;
  if (idx >= total) return;
  int e = idx / Kf;
  int f = idx - e * Kf;
  atomicAdd(&agg[(size_t)dst[e] * Kf + f], hin[(size_t)src[e] * Kf + f]);
}

// ---------------- edge scatter-add, 128 features, float4 vectorized ----------------
__global__ void k_edge_scatter128(const float* __restrict__ hin,
                                  const int* __restrict__ src,
                                  const int* __restrict__ dst,
                                  float* __restrict__ agg,
                                  int E) {
  int idx = blockIdx.x * blockDim.x + threadIdx.x;
  if (idx >= E * 32) return;
  int e = idx >> 5;
  int f = (idx & 31) << 2;
  const float4 v = *(const float4*)(hin + (size_t)src[e] * HDIM + f);
  float* o = agg + (size_t)dst[e] * HDIM + f;
  atomicAdd(o + 0, v.x);
  atomicAdd(o + 1, v.y);
  atomicAdd(o + 2, v.z);
  atomicAdd(o + 3, v.w);
}

// ---------------- fused dual-GEMM GCN layer via V_WMMA_F32_16X16X4_F32 ----------------
// out[n][j] = relu(sum_k A1[n][k]*W[k][j] + b[j]) + relu(sum_k A2[n][k]*Wr[k][j] + br[j])
// block = 256 threads (8 waves); wave w covers output columns [16w, 16w+16);
// block x covers rows [16*bx, 16*bx+16).
// Branch-free inner loop: A rows are index-clamped (garbage rows never stored),
// K tail handled in a single select-based step.
__global__ __launch_bounds__(256) void k_gcn_gemm(
    const float* __restrict__ A1, const float* __restrict__ A2,
    const float* __restrict__ W, const float* __restrict__ bb,
    const float* __restrict__ Wr, const float* __restrict__ br,
    float* __restrict__ out, int N, int K) {
  const int wave = threadIdx.x >> 5;
  const int lane = threadIdx.x & 31;
  const int rowBase = blockIdx.x << 4;
  const int colBase = wave << 4;
  const int r = lane & 15;
  const int rowC = min(rowBase + r, N - 1);  // clamped: OOB rows only feed unstored C rows
  const int col = colBase + r;
  const int kOff = (lane >> 4) << 1;  // lanes 0-15: K pair (+0,+1); lanes 16-31: (+2,+3)

  v8f acc1 = {0.f, 0.f, 0.f, 0.f, 0.f, 0.f, 0.f, 0.f};
  v8f acc2 = {0.f, 0.f, 0.f, 0.f, 0.f, 0.f, 0.f, 0.f};

  const float* a1p = A1 + (size_t)rowC * K + kOff;
  const float* a2p = A2 + (size_t)rowC * K + kOff;
  const float* wp  = W  + (size_t)kOff * HDIM + col;
  const float* wrp = Wr + (size_t)kOff * HDIM + col;

  const int Kmain = K & ~3;
  int kb = 0;
#pragma unroll 2
  for (; kb < Kmain; kb += 4) {
    const v2f a1 = *(const v2f*)(a1p + kb);
    const v2f a2 = *(const v2f*)(a2p + kb);
    v2f b1, b2;
    b1.x = wp[(size_t)kb * HDIM];
    b1.y = wp[(size_t)kb * HDIM + HDIM];
    b2.x = wrp[(size_t)kb * HDIM];
    b2.y = wrp[(size_t)kb * HDIM + HDIM];
    acc1 = __builtin_amdgcn_wmma_f32_16x16x4_f32(false, a1, false, b1,
                                                 (short)0, acc1, false, false);
    acc2 = __builtin_amdgcn_wmma_f32_16x16x4_f32(false, a2, false, b2,
                                                 (short)0, acc2, false, false);
  }
  if (Kmain < K) {  // tail (K % 4 != 0, e.g. K=74): clamped loads + value selects
    const int k0 = kb + kOff;
    const int k1 = k0 + 1;
    const int k0c = min(k0, K - 1);
    const int k1c = min(k1, K - 1);
    const size_t rB = (size_t)rowC * K;
    v2f a1, a2, b1, b2;
    a1.x = A1[rB + k0c];
    a1.y = A1[rB + k1c];
    a2.x = A2[rB + k0c];
    a2.y = A2[rB + k1c];
    const float w0  = W[(size_t)k0c * HDIM + col];
    const float w1  = W[(size_t)k1c * HDIM + col];
    const float wr0 = Wr[(size_t)k0c * HDIM + col];
    const float wr1 = Wr[(size_t)k1c * HDIM + col];
    b1.x = (k0 < K) ? w0 : 0.0f;   // zeroing B makes garbage A tail harmless
    b1.y = (k1 < K) ? w1 : 0.0f;
    b2.x = (k0 < K) ? wr0 : 0.0f;
    b2.y = (k1 < K) ? wr1 : 0.0f;
    acc1 = __builtin_amdgcn_wmma_f32_16x16x4_f32(false, a1, false, b1,
                                                 (short)0, acc1, false, false);
    acc2 = __builtin_amdgcn_wmma_f32_16x16x4_f32(false, a2, false, b2,
                                                 (short)0, acc2, false, false);
  }

  const float bc = bb[col];
  const float brc = br[col];
  const int mHi = (lane >> 4) << 3;  // C/D: vgpr i holds row i (lanes<16) or i+8 (lanes>=16)
#pragma unroll
  for (int i = 0; i < 8; ++i) {
    const int orow = rowBase + mHi + i;
    if (orow < N) {
      const float v = fmaxf(acc1[i] + bc, 0.0f) + fmaxf(acc2[i] + brc, 0.0f);
      out[(size_t)orow * HDIM + col] = v;
    }
  }
}

// ---------------- weighted-sum + max readout ----------------
// hg[g][0:128] = sum_n sigmoid(h2[n].w_atom + b)*h2[n];  hg[g][128:256] = max_n h2[n]
// h2 >= 0 (sum of two ReLUs) so integer atomicMax on float bits is exact with init 0.
__global__ __launch_bounds__(128) void k_readout(
    const float* __restrict__ h2, const int* __restrict__ n2g,
    const float* __restrict__ w_atom, const float* __restrict__ b_atom,
    float* __restrict__ hg, int N) {
  const int n = blockIdx.x;
  const int f = threadIdx.x;
  __shared__ float s[128];
  const float v = h2[(size_t)n * HDIM + f];
  s[f] = v * w_atom[f];
  __syncthreads();
  for (int off = 64; off > 0; off >>= 1) {
    if (f < off) s[f] += s[f + off];
    __syncthreads();
  }
  const float wgt = 1.0f / (1.0f + expf(-(s[0] + b_atom[0])));
  const int g = n2g[n];
  atomicAdd(&hg[(size_t)g * 256 + f], wgt * v);
  atomicMax((int*)&hg[(size_t)g * 256 + 128 + f], __float_as_int(v));
}

// ---------------- MLP head: Linear->ReLU->BN(affine)->Linear->sigmoid ----------------
__global__ __launch_bounds__(128) void k_mlp(
    const float* __restrict__ hg, const float* __restrict__ Wc1,
    const float* __restrict__ bc1, const float* __restrict__ gma,
    const float* __restrict__ bta, const float* __restrict__ Wc2,
    const float* __restrict__ bc2, float* __restrict__ out) {
  const int g = blockIdx.x;
  const int j = threadIdx.x;
  __shared__ float sh[256];
  __shared__ float s2[128];
  sh[j] = hg[(size_t)g * 256 + j];
  sh[j + 128] = hg[(size_t)g * 256 + 128 + j];
  __syncthreads();
  float acc = bc1[j];
#pragma unroll 8
  for (int k = 0; k < 256; ++k) acc = fmaf(sh[k], Wc1[k * HDIM + j], acc);
  const float z = fmaxf(acc, 0.0f) * gma[j] + bta[j];
  s2[j] = z * Wc2[j];
  __syncthreads();
  for (int off = 64; off > 0; off >>= 1) {
    if (j < off) s2[j] += s2[j + off];
    __syncthreads();
  }
  if (j == 0) out[g] = 1.0f / (1.0f + expf(-(s2[0] + bc2[0])));
}

extern "C" void kernel_launch(void* const* d_in, const int* in_sizes, int n_in,
                              void* d_out, int out_size, void* d_ws, size_t ws_size,
                              hipStream_t stream) {
  (void)n_in; (void)ws_size;
  const float* h      = (const float*)d_in[0];
  const int*   esrc   = (const int*)d_in[1];
  const int*   edst   = (const int*)d_in[2];
  const int*   n2g    = (const int*)d_in[3];
  const float* W0     = (const float*)d_in[4];
  const float* b0     = (const float*)d_in[5];
  const float* Wr0    = (const float*)d_in[6];
  const float* br0    = (const float*)d_in[7];
  const float* W1     = (const float*)d_in[8];
  const float* b1     = (const float*)d_in[9];
  const float* Wr1    = (const float*)d_in[10];
  const float* br1    = (const float*)d_in[11];
  const float* w_atom = (const float*)d_in[12];
  const float* b_atom = (const float*)d_in[13];
  const float* Wc1    = (const float*)d_in[14];
  const float* bc1    = (const float*)d_in[15];
  const float* gma    = (const float*)d_in[16];
  const float* bta    = (const float*)d_in[17];
  const float* Wc2    = (const float*)d_in[18];
  const float* bc2    = (const float*)d_in[19];
  float* out = (float*)d_out;

  const int E = in_sizes[1];
  const int N = in_sizes[3];
  const int F = in_sizes[0] / N;  // 74
  const int G = out_size;         // 2000

  float* agg = (float*)d_ws;              // N*128 floats (reused both layers)
  float* h1  = agg + (size_t)N * HDIM;    // N*128
  float* h2  = h1 + (size_t)N * HDIM;     // N*128
  float* hg  = h2 + (size_t)N * HDIM;     // G*256

  const int TB = 256;

  // ---- layer 0 ----
  {
    const int n = N * F;
    k_zero<<<(n + TB - 1) / TB, TB, 0, stream>>>(agg, n);
    const int tot = E * F;
    k_edge_scatter<<<(tot + TB - 1) / TB, TB, 0, stream>>>(h, esrc, edst, agg, tot, F);
    k_gcn_gemm<<<(N + 15) / 16, 256, 0, stream>>>(agg, h, W0, b0, Wr0, br0, h1, N, F);
  }
  // ---- layer 1 ----
  {
    const int n = N * HDIM;
    k_zero<<<(n + TB - 1) / TB, TB, 0, stream>>>(agg, n);
    const int tot = E * 32;
    k_edge_scatter128<<<(tot + TB - 1) / TB, TB, 0, stream>>>(h1, esrc, edst, agg, E);
    k_gcn_gemm<<<(N + 15) / 16, 256, 0, stream>>>(agg, h1, W1, b1, Wr1, br1, h2, N, HDIM);
  }
  // ---- readout + MLP ----
  {
    const int n = G * 256;
    k_zero<<<(n + TB - 1) / TB, TB, 0, stream>>>(hg, n);
    k_readout<<<N, 128, 0, stream>>>(h2, n2g, w_atom, b_atom, hg, N);
    k_mlp<<<G, 128, 0, stream>>>(hg, Wc1, bc1, gma, bta, Wc2, bc2, out);
  }
}